// SocialPooling_83099027243706
// MI455X (gfx1250) — compile-verified
//
#include <hip/hip_runtime.h>
#include <hip/hip_bf16.h>
#include <math.h>

// Problem constants (match reference)
#define SS   32
#define NN   128
#define HH   64
#define GG   8
#define OUTD 64
#define EPSV 1e-5f

// Pooling tiling: 16 cells per pass -> 4 passes, ~99 KB static LDS (probe of WGP LDS cap)
#define CB    16
#define NPASS ((GG*GG)/CB)   // 4 passes
#define PROW  (CB*HH + 4)    // padded pool row stride (floats): 1028 -> 4-bank lane skew

typedef float v2f __attribute__((ext_vector_type(2)));
typedef float v8f __attribute__((ext_vector_type(8)));

// -------- Kernel 1: fused social pooling + GEMM (f32 WMMA), writes pre-BN x --------
__global__ __launch_bounds__(128)
void sp_pool_gemm(const float* __restrict__ hs_g,
                  const float* __restrict__ pos_g,
                  const float* __restrict__ W,
                  const float* __restrict__ b,
                  const int*   __restrict__ sse,
                  float*       __restrict__ xout)
{
    __shared__ __align__(16) float hs[NN][HH];            // 32 KB
    __shared__ float px[NN], py[NN];                      // 1 KB
    __shared__ unsigned char cellmat[16][NN];             // 2 KB
    __shared__ __align__(16) float pool[16][PROW];        // ~64.3 KB

    const int wg    = blockIdx.x;
    const int frame = wg >> 3;          // 32 frames
    const int i0    = (wg & 7) << 4;    // 8 tiles of 16 pedestrians
    const int t     = threadIdx.x;
    const int start = sse[frame * 2];

    // Load frame hidden states (128x64 f32) cooperatively as float4
    {
        const float4* src = (const float4*)(hs_g + (size_t)start * HH);
        float4* dst = (float4*)(&hs[0][0]);
        for (int k = t; k < NN * HH / 4; k += 128) dst[k] = src[k];
    }
    if (t < NN) {
        px[t] = pos_g[(size_t)(start + t) * 2 + 0];
        py[t] = pos_g[(size_t)(start + t) * 2 + 1];
    }
    __syncthreads();

    // Cell assignment matrix for this i-tile (replicates reference masking)
    for (int e = t; e < 16 * NN; e += 128) {
        const int ii = e >> 7;
        const int j  = e & 127;
        const int i  = i0 + ii;
        const float xi = px[i], yi = py[i];
        const float tlx = xi - 1.0f, tly = yi + 1.0f;
        const float brx = xi + 1.0f, bry = yi - 1.0f;
        const float xj = px[j], yj = py[j];
        unsigned char c = 255u;
        const bool bad = (xj <= tlx) | (yj >= tly) | (xj >= brx) | (yj <= bry) | (i == j);
        if (!bad) {
            const int gx = (int)floorf((xj - tlx) / 2.0f * 8.0f);
            const int gy = (int)floorf((tly - yj) / 2.0f * 8.0f);
            c = (unsigned char)(gy * GG + gx);
        }
        cellmat[ii][j] = c;
    }

    // WMMA lane mapping (V_WMMA_F32_16X16X4_F32)
    // A (16x4): lanes 0-15 -> M=lane, VGPR0 holds K=0 (lo lanes)/K=2 (hi lanes), VGPR1 K=1/K=3
    // B (4x16): lanes 0-15 -> N=lane, same K interleave
    // C/D (16x16): VGPR r -> M=r (lo lanes) / M=r+8 (hi lanes), N=lane&15
    const int wave = t >> 5;
    const int lane = t & 31;
    const int n0   = wave << 4;             // each of 4 waves owns a 16-col n-tile
    const int Mrow = lane & 15;
    const int Ncol = lane & 15;
    const int ka   = (lane < 16) ? 0 : 2;
    v8f acc = {};

    const int il = t >> 3;                  // scatter: thread owns (i_local, 8-col slice)
    const int hg = (t & 7) << 3;

    for (int pass = 0; pass < NPASS; ++pass) {
        __syncthreads();
        // zero pool tile
        for (int k = t; k < 16 * PROW; k += 128) ((float*)pool)[k] = 0.0f;
        __syncthreads();

        // scatter-add: conflict-free, each thread owns pool[il][*][hg..hg+7]
        {
            const int clo = pass * CB;
            for (int j = 0; j < NN; ++j) {
                const int c = (int)cellmat[il][j] - clo;
                if ((unsigned)c < (unsigned)CB) {
                    float* p = &pool[il][c * HH + hg];
                    const float* h = &hs[j][hg];
                    float4 p0 = *(float4*)p,       p1 = *(float4*)(p + 4);
                    float4 h0 = *(const float4*)h, h1 = *(const float4*)(h + 4);
                    p0.x += h0.x; p0.y += h0.y; p0.z += h0.z; p0.w += h0.w;
                    p1.x += h1.x; p1.y += h1.y; p1.z += h1.z; p1.w += h1.w;
                    *(float4*)p = p0; *(float4*)(p + 4) = p1;
                }
            }
        }
        __syncthreads();

        // GEMM over this pass's K-block (CB cells * 64)
        for (int c = 0; c < CB; ++c) {
            const int cellg = pass * CB + c;
            const float* Wrow = W + (size_t)cellg * HH * OUTD;
            const float* Arow = &pool[Mrow][c * HH];
#pragma unroll
            for (int kk = 0; kk < HH; kk += 4) {
                v2f A, B;
                A.x = Arow[kk + ka];
                A.y = Arow[kk + ka + 1];
                B.x = Wrow[(size_t)(kk + ka) * OUTD + n0 + Ncol];
                B.y = Wrow[(size_t)(kk + ka + 1) * OUTD + n0 + Ncol];
                acc = __builtin_amdgcn_wmma_f32_16x16x4_f32(
                    false, A, false, B, (short)0, acc, false, false);
            }
        }
    }

    // epilogue: x = acc + bias  (BN handled by later kernels)
    {
        const int gi  = frame * NN + i0;
        const int col = n0 + Ncol;
        const float bias = b[col];
#pragma unroll
        for (int r = 0; r < 8; ++r) {
            const int M = r + ((lane >> 4) << 3);
            xout[(size_t)(gi + M) * OUTD + col] = acc[r] + bias;
        }
    }
}

// -------- Kernel 2: per-column batch stats (mean, invstd), 1024 threads --------
__global__ __launch_bounds__(1024)
void sp_stats(const float* __restrict__ x, float* __restrict__ stats)
{
    __shared__ float ssum[16][OUTD];
    __shared__ float ssq[16][OUTD];
    const int t = threadIdx.x;
    const int col = t & 63;
    const int part = t >> 6;          // 16 partials of 256 rows each
    float s = 0.0f, q = 0.0f;
    const int r0 = part * 256;
    for (int r = r0; r < r0 + 256; ++r) {
        const float v = x[(size_t)r * OUTD + col];
        s += v; q += v * v;
    }
    ssum[part][col] = s;
    ssq[part][col]  = q;
    __syncthreads();
    if (part == 0) {
        float st = 0.0f, qt = 0.0f;
#pragma unroll
        for (int p = 0; p < 16; ++p) { st += ssum[p][col]; qt += ssq[p][col]; }
        const float mean = st * (1.0f / 4096.0f);
        float var = qt * (1.0f / 4096.0f) - mean * mean;
        var = fmaxf(var, 0.0f);
        stats[col]        = mean;
        stats[OUTD + col] = rsqrtf(var + EPSV);
    }
}

// -------- Kernel 3: BN apply + ReLU, in place on d_out --------
__global__ __launch_bounds__(256)
void sp_bn_relu(float* __restrict__ x, const float* __restrict__ stats,
                const float* __restrict__ gamma, const float* __restrict__ beta)
{
    const int idx = blockIdx.x * 256 + threadIdx.x;   // 65536 float4's
    float4* x4 = (float4*)x;
    float4 v = x4[idx];
    const int c0 = (idx & 15) * 4;
    float m, is, g, be;
    m = stats[c0 + 0]; is = stats[OUTD + c0 + 0]; g = gamma[c0 + 0]; be = beta[c0 + 0];
    v.x = fmaxf((v.x - m) * is * g + be, 0.0f);
    m = stats[c0 + 1]; is = stats[OUTD + c0 + 1]; g = gamma[c0 + 1]; be = beta[c0 + 1];
    v.y = fmaxf((v.y - m) * is * g + be, 0.0f);
    m = stats[c0 + 2]; is = stats[OUTD + c0 + 2]; g = gamma[c0 + 2]; be = beta[c0 + 2];
    v.z = fmaxf((v.z - m) * is * g + be, 0.0f);
    m = stats[c0 + 3]; is = stats[OUTD + c0 + 3]; g = gamma[c0 + 3]; be = beta[c0 + 3];
    v.w = fmaxf((v.w - m) * is * g + be, 0.0f);
    x4[idx] = v;
}

extern "C" void kernel_launch(void* const* d_in, const int* in_sizes, int n_in,
                              void* d_out, int out_size, void* d_ws, size_t ws_size,
                              hipStream_t stream)
{
    const float* hs    = (const float*)d_in[0];   // [4096,64]
    const float* pos   = (const float*)d_in[1];   // [4096,2]
    const float* W     = (const float*)d_in[2];   // [4096,64]
    const float* b     = (const float*)d_in[3];   // [64]
    const float* gamma = (const float*)d_in[4];   // [64]
    const float* beta  = (const float*)d_in[5];   // [64]
    const int*   sse   = (const int*)d_in[6];     // [32,2]
    float* out   = (float*)d_out;                 // [4096,64]
    float* stats = (float*)d_ws;                  // mean[64], invstd[64]

    sp_pool_gemm<<<SS * 8, 128, 0, stream>>>(hs, pos, W, b, sse, out);
    sp_stats<<<1, 1024, 0, stream>>>(out, stats);
    sp_bn_relu<<<(SS * NN * OUTD / 4) / 256, 256, 0, stream>>>(out, stats, gamma, beta);
}